// SineGen_73607149519381
// MI455X (gfx1250) — compile-verified
//
#include <hip/hip_runtime.h>

typedef __attribute__((ext_vector_type(2))) float v2f;
typedef __attribute__((ext_vector_type(4))) float v4f;
typedef __attribute__((ext_vector_type(8))) float v8f;

#define B_N   8
#define L_N   8192
#define UPP_N 320
#define TWO_PI 6.28318530717958647692f

// factor[l]: 0.1 at l=0, 0.1/(2(l+1)) for 1..8, else 1.  All constants fold at
// compile time to the exact f32 values the reference's f32 divide produces.
__device__ __forceinline__ float factor_of(int l) {
  float fac = 1.0f;
  fac = (l == 8) ? (0.1f / 18.0f) : fac;
  fac = (l == 7) ? (0.1f / 16.0f) : fac;
  fac = (l == 6) ? (0.1f / 14.0f) : fac;
  fac = (l == 5) ? (0.1f / 12.0f) : fac;
  fac = (l == 4) ? (0.1f / 10.0f) : fac;
  fac = (l == 3) ? (0.1f / 8.0f)  : fac;
  fac = (l == 2) ? (0.1f / 6.0f)  : fac;
  fac = (l == 1) ? (0.1f / 4.0f)  : fac;
  fac = (l == 0) ? 0.1f           : fac;
  return fac;
}

// weighted sine term: sin(2*pi*(frac(f0*89/16000) + rand_ini)) * factor[l]
// Branch-free: callers pass a clamped (always-valid) row index.
__device__ __forceinline__ float weighted_sin(const float* __restrict__ f0,
                                              const float* __restrict__ ri,
                                              int m, int l) {
  float x    = f0[m * L_N + l] * (89.0f / 16000.0f);
  float frac = x - floorf(x);
  float r    = (l == 0) ? 0.0f : ri[m * L_N + l];
  return sinf(TWO_PI * (frac + r)) * factor_of(l);
}

// ---------------------------------------------------------------------------
// Stage A1: per-batch weighted-sine dot product via V_WMMA_F32_16X16X4_F32.
// A (16x4, documented layout: V0 lanes0-15 K=0 / lanes16-31 K=2; V1 K=1/K=3)
// holds sin*factor for batch m = lane&15 (rows 8..15 masked to zero with a
// select, not divergent branches).  B = all-ones, so D[m,n] = sum_k A[m,k]
// regardless of B's row layout.  64 blocks x 8 waves; deterministic combine.
// ---------------------------------------------------------------------------
__global__ void __launch_bounds__(256)
sine_dot_wmma(const float* __restrict__ f0, const float* __restrict__ ri,
              float* __restrict__ partial /* [64][8] */) {
  __shared__ float wpart[8][8];
  const int tid  = threadIdx.x;
  const int wave = tid >> 5;
  const int lane = tid & 31;
  const int m    = lane & 15;
  const int mc   = (m < B_N) ? m : (B_N - 1);  // clamped: loads always valid
  const float mmask = (m < B_N) ? 1.0f : 0.0f;
  const int kh   = lane >> 4;

  v8f c    = {0.f, 0.f, 0.f, 0.f, 0.f, 0.f, 0.f, 0.f};
  v2f ones = {1.0f, 1.0f};

  const int wglobal = blockIdx.x * 8 + wave;
#pragma unroll 1
  for (int chunk = wglobal; chunk < (L_N / 4); chunk += 64 * 8) {
    const int l0 = chunk * 4 + 2 * kh;
    v2f a;
    a.x = mmask * weighted_sin(f0, ri, mc, l0);
    a.y = mmask * weighted_sin(f0, ri, mc, l0 + 1);
    // 8 args: (neg_a, A, neg_b, B, c_mod, C, reuse_a, reuse_b)
    c = __builtin_amdgcn_wmma_f32_16x16x4_f32(false, a, false, ones,
                                              (short)0, c, false, false);
  }

  // C layout: VGPR j, lanes 0-15 => M=j, N=lane.  All N columns equal
  // (B uniform), so lane 0 holds wave-partials for batches j=0..7 in c[0..7].
  if (lane == 0) {
#pragma unroll
    for (int j = 0; j < 8; ++j) wpart[wave][j] = c[j];
  }
  __syncthreads();
  if (tid < B_N) {
    float s = 0.0f;
#pragma unroll
    for (int w = 0; w < 8; ++w) s += wpart[w][tid];
    partial[blockIdx.x * 8 + tid] = s;
  }
}

// Stage A2: deterministic fixed-order reduction of 64 block partials.
__global__ void __launch_bounds__(32)
reduce_final(const float* __restrict__ partial, float* __restrict__ S) {
  const int tid = threadIdx.x;
  if (tid < B_N) {
    float s = 0.0f;
    for (int blk = 0; blk < 64; ++blk) s += partial[blk * 8 + tid];
    S[tid] = s;
  }
}

// ---------------------------------------------------------------------------
// Stage B: bandwidth kernel (sets the runtime: ~335 MB @ 23.3 TB/s ~ 14 us).
// One block per (b, t) row of 8192 floats.  noise_z row staged to LDS with
// global_load_async_to_lds_b128 (ASYNCcnt; no VGPR destinations), outputs
// written with non-temporal B128 stores (never re-read).
// ---------------------------------------------------------------------------
__global__ void __launch_bounds__(256)
sine_stream(const v4f* __restrict__ z, const float* __restrict__ f0,
            const float* __restrict__ S, float* __restrict__ out) {
  __shared__ v4f buf[L_N / 4];  // 32 KB
  const int tid = threadIdx.x;
  const int row = blockIdx.x;           // row = b*UPP + t
  const int b   = row / UPP_N;
  const int t   = row - b * UPP_N;
  const size_t rowbase = (size_t)row * (L_N / 4);  // in float4 units

  // Async-stage this row's noise into LDS (8 x B128 per thread).
#pragma unroll
  for (int i = 0; i < 8; ++i) {
    const int idx = tid + i * 256;
    const unsigned lds = (unsigned)(size_t)(&buf[idx]);  // low 32b = LDS offset
    const unsigned long long gaddr =
        (unsigned long long)(z + rowbase + idx);
    asm volatile("global_load_async_to_lds_b128 %0, %1, off"
                 :
                 : "v"(lds), "v"(gaddr)
                 : "memory");
  }
  asm volatile("s_wait_asynccnt 0" ::: "memory");
  // Each thread reads back only the LDS slots it filled -> no barrier needed.

  const size_t N = (size_t)B_N * UPP_N * L_N;
  v4f* __restrict__ out_s = (v4f*)out;
  v4f* __restrict__ out_u = (v4f*)(out + N);
  v4f* __restrict__ out_n = (v4f*)(out + 2 * N);

  if (t == 0) {
    const float sb = S[b] * 0.1f;  // sines * SINE_AMP
    const v4f* f4  = (const v4f*)f0 + (size_t)b * (L_N / 4);
#pragma unroll
    for (int i = 0; i < 8; ++i) {
      const int idx = tid + i * 256;
      v4f zv = buf[idx];
      v4f fv = f4[idx];
      v4f uv, nz, sn;
#pragma unroll
      for (int k = 0; k < 4; ++k) {
        const float u  = (fv[k] > 0.0f) ? 1.0f : 0.0f;
        const float na = (u > 0.0f) ? 0.003f : 0.1f;
        const float n  = na * zv[k];
        uv[k] = u;
        nz[k] = n;
        sn[k] = sb * u + n;
      }
      const size_t o = rowbase + idx;
      __builtin_nontemporal_store(sn, out_s + o);
      __builtin_nontemporal_store(uv, out_u + o);
      __builtin_nontemporal_store(nz, out_n + o);
    }
  } else {
    // t > 0: uv = 0, noise = 0.1*z, sine_wavs = noise.
#pragma unroll
    for (int i = 0; i < 8; ++i) {
      const int idx = tid + i * 256;
      v4f zv   = buf[idx];
      v4f nz   = zv * 0.1f;
      v4f zero = {0.f, 0.f, 0.f, 0.f};
      const size_t o = rowbase + idx;
      __builtin_nontemporal_store(nz, out_s + o);
      __builtin_nontemporal_store(zero, out_u + o);
      __builtin_nontemporal_store(nz, out_n + o);
    }
  }
}

extern "C" void kernel_launch(void* const* d_in, const int* in_sizes, int n_in,
                              void* d_out, int out_size, void* d_ws,
                              size_t ws_size, hipStream_t stream) {
  const float* f0 = (const float*)d_in[0];  // (8, 8192)
  const float* ri = (const float*)d_in[1];  // (8, 8192)
  const float* z  = (const float*)d_in[2];  // (8, 320, 8192)
  // d_in[3] = upp (compile-time constant 320), unused.

  float* partial = (float*)d_ws;       // 64*8 floats
  float* S       = partial + 64 * 8;   // 8 floats

  sine_dot_wmma<<<64, 256, 0, stream>>>(f0, ri, partial);
  reduce_final<<<1, 32, 0, stream>>>(partial, S);
  sine_stream<<<B_N * UPP_N, 256, 0, stream>>>((const v4f*)z, f0, S,
                                               (float*)d_out);

  (void)in_sizes;
  (void)n_in;
  (void)out_size;
  (void)ws_size;
}